// Equalization_3624952398041
// MI455X (gfx1250) — compile-verified
//
#include <hip/hip_runtime.h>
#include <hip/hip_bf16.h>

// Histogram equalization, [B=64, H=512, W=512, C=3] int32 in [0,255].
// Memory-bound: input read 2x + output write 1x. 192 MB L2 vs 201 MB input:
// keep pass-1 reads RT (fill L2), make pass-2 reads/writes non-temporal so
// the streaming store traffic doesn't evict the input before its 2nd read.
// Key fact: bin index == pixel value for v in [0,255] (floor(v*256/255)==v).

#define IMG_H 512
#define IMG_W 512
#define IMG_C 3
#define IMG_ELEMS (IMG_H * IMG_W * IMG_C) /* 786432 */
#define HIST_BLOCKS_PER_IMG 24
#define GATHER_BLOCKS_PER_IMG 96

typedef int vint4 __attribute__((ext_vector_type(4)));

static __device__ __forceinline__ int clamp255(int v) {
    v = v < 0 ? 0 : v;
    return v > 255 ? 255 : v;
}

__global__ __launch_bounds__(256) void eq_zero_hist(int* __restrict__ histg, int n) {
    int i = blockIdx.x * 256 + threadIdx.x;
    if (i < n) histg[i] = 0;
}

// One block = one contiguous stripe of one image. 3 LDS histograms (3 KB),
// flushed once with global atomics. Regular-temporal loads: prime L2 for the
// gather pass re-read.
__global__ __launch_bounds__(256) void eq_hist(const int* __restrict__ img,
                                               int* __restrict__ histg) {
    __shared__ int lh[3 * 256];
    const int t = threadIdx.x;
    const int b = blockIdx.y;
    lh[t] = 0; lh[t + 256] = 0; lh[t + 512] = 0;
    __syncthreads();

    const vint4* __restrict__ ip = (const vint4*)(img + (size_t)b * IMG_ELEMS);
    const int vec_total = IMG_ELEMS / 4; /* 196608 */
    const int per_block = (vec_total + gridDim.x - 1) / gridDim.x;
    const int base = blockIdx.x * per_block;
    const int lim  = (base + per_block < vec_total) ? base + per_block : vec_total;

    for (int vi = base + t; vi < lim; vi += 256) {
        if (vi + 256 < lim) __builtin_prefetch(&ip[vi + 256], 0, 0); // global_prefetch_b8
        vint4 v = ip[vi];
        int c0 = (vi * 4) % 3;
        int c1 = c0 + 1; if (c1 == 3) c1 = 0;
        int c2 = c1 + 1; if (c2 == 3) c2 = 0;
        int c3 = c2 + 1; if (c3 == 3) c3 = 0;
        atomicAdd(&lh[c0 * 256 + clamp255(v.x)], 1);
        atomicAdd(&lh[c1 * 256 + clamp255(v.y)], 1);
        atomicAdd(&lh[c2 * 256 + clamp255(v.z)], 1);
        atomicAdd(&lh[c3 * 256 + clamp255(v.w)], 1);
    }
    __syncthreads();

    int* gh = histg + (size_t)b * 768;
    if (lh[t])       atomicAdd(&gh[t],       lh[t]);
    if (lh[t + 256]) atomicAdd(&gh[t + 256], lh[t + 256]);
    if (lh[t + 512]) atomicAdd(&gh[t + 512], lh[t + 512]);
}

// One block per (image, channel) task: inclusive scan + last-nonzero-bin
// reduction + shifted-cumsum LUT, exactly mirroring the reference math.
__global__ __launch_bounds__(256) void eq_lut(const int* __restrict__ histg,
                                              int* __restrict__ lutg) {
    __shared__ int h[256];
    __shared__ int cs[256];
    __shared__ int red[256];
    const int t = threadIdx.x;
    const int task = blockIdx.x;

    int hv = histg[(size_t)task * 256 + t];
    h[t] = hv;
    cs[t] = hv;
    red[t] = (hv > 0) ? t : -1;
    __syncthreads();

    // Hillis-Steele inclusive scan over 256 bins
    for (int off = 1; off < 256; off <<= 1) {
        int v = (t >= off) ? cs[t - off] : 0;
        __syncthreads();
        cs[t] += v;
        __syncthreads();
    }
    // max-reduce: index of last nonzero histogram bin
    for (int s = 128; s > 0; s >>= 1) {
        if (t < s) red[t] = max(red[t], red[t + s]);
        __syncthreads();
    }

    const int total     = cs[255];
    const int last_idx  = red[0];
    const int last_val  = (last_idx >= 0) ? h[last_idx] : 0;
    const int step      = (total - last_val) / 255;       // BINS-1
    const int step_safe = (step == 0) ? 1 : step;

    // lut = concat([0], ((cumsum + step//2) // step)[:-1]), clipped to [0,255]
    int raw = (t == 0) ? 0 : (cs[t - 1] + (step_safe >> 1)) / step_safe;
    int lv  = clamp255(raw);
    lutg[(size_t)task * 256 + t] = (step == 0) ? t : lv;  // step==0 -> identity
}

// Streaming translate pass. The 3 KB of per-image LUTs are staged into LDS
// with CDNA5 async global->LDS B128 copies (ASYNCcnt), then fenced.
// Images processed in reverse dispatch order to hit the freshest L2 lines
// left behind by the histogram pass; all pass-2 traffic is non-temporal.
__global__ __launch_bounds__(256) void eq_gather(const int* __restrict__ img,
                                                 const int* __restrict__ lutg,
                                                 int* __restrict__ out) {
    __shared__ int lut3[3 * 256];
    const int t = threadIdx.x;
    const int b = (int)(gridDim.y - 1u - blockIdx.y);   // reverse image order

    if (t < 192) {
        // Flat LDS pointer: addr[31:0] is the wave-relative LDS byte offset
        // (hardware truncates upper bits), which is what VDST must hold.
        unsigned lds_off = (unsigned)(unsigned long long)(&lut3[t * 4]);
        unsigned long long ga =
            (unsigned long long)(lutg + (size_t)b * 768 + (size_t)t * 4);
        asm volatile("global_load_async_to_lds_b128 %0, %1, off"
                     :: "v"(lds_off), "v"(ga)
                     : "memory");
    }
    asm volatile("s_wait_asynccnt 0x0" ::: "memory");
    __syncthreads();

    const vint4* __restrict__ ip = (const vint4*)(img + (size_t)b * IMG_ELEMS);
    vint4* __restrict__ op = (vint4*)(out + (size_t)b * IMG_ELEMS);
    const int vec_total = IMG_ELEMS / 4;
    const int stride = gridDim.x * 256;

    for (int vi = blockIdx.x * 256 + t; vi < vec_total; vi += stride) {
        if (vi + stride < vec_total) __builtin_prefetch(&ip[vi + stride], 0, 0);
        vint4 v = __builtin_nontemporal_load(&ip[vi]);   // last use of input line
        int c0 = (vi * 4) % 3;
        int c1 = c0 + 1; if (c1 == 3) c1 = 0;
        int c2 = c1 + 1; if (c2 == 3) c2 = 0;
        int c3 = c2 + 1; if (c3 == 3) c3 = 0;
        vint4 r;
        r.x = lut3[c0 * 256 + clamp255(v.x)];
        r.y = lut3[c1 * 256 + clamp255(v.y)];
        r.z = lut3[c2 * 256 + clamp255(v.z)];
        r.w = lut3[c3 * 256 + clamp255(v.w)];
        __builtin_nontemporal_store(r, &op[vi]);         // never re-read
    }
}

extern "C" void kernel_launch(void* const* d_in, const int* in_sizes, int n_in,
                              void* d_out, int out_size, void* d_ws, size_t ws_size,
                              hipStream_t stream) {
    const int* img = (const int*)d_in[0];
    int* out = (int*)d_out;

    const int n = in_sizes[0];              // 64*512*512*3
    const int B = n / IMG_ELEMS;            // 64

    int* histg = (int*)d_ws;                // [B*3, 256] int
    int* lutg  = histg + (size_t)B * 768;   // [B*3, 256] int

    const int hist_elems = B * 768;
    eq_zero_hist<<<(hist_elems + 255) / 256, 256, 0, stream>>>(histg, hist_elems);
    eq_hist<<<dim3(HIST_BLOCKS_PER_IMG, B), 256, 0, stream>>>(img, histg);
    eq_lut<<<dim3(B * 3), 256, 0, stream>>>(histg, lutg);
    eq_gather<<<dim3(GATHER_BLOCKS_PER_IMG, B), 256, 0, stream>>>(img, lutg, out);
}